// MultilevelDense_32512902431065
// MI455X (gfx1250) — compile-verified
//
#include <hip/hip_runtime.h>

typedef __attribute__((ext_vector_type(2))) float v2f;
typedef __attribute__((ext_vector_type(4))) float v4f;
typedef __attribute__((ext_vector_type(8))) float v8f;

#define DD 1024
#define BB 64

// softplus(x) = max(x,0) + ln2 * log2(1 + exp2(-|x| * log2e))
// Uses raw v_exp_f32 / v_log_f32 (2 TRANS + ~4 VALU) instead of the
// range-checked __expf/__logf expansions (~25 VALU each).
__device__ __forceinline__ float softplus_fast(float x) {
    float t = __builtin_amdgcn_exp2f(fabsf(x) * -1.442695040888963f);
    float l = __builtin_amdgcn_logf(1.0f + t);          // log2(1+t)
    return fmaf(0.6931471805599453f, l, fmaxf(x, 0.0f));
}

// One block = one sample b + 8 u-tiles (one 16-row tile per wave).
// Each wave accumulates out[b, u0:u0+16] with V_WMMA_F32_16X16X4_F32:
//   A[m,k] = w_samp[b, u0+m, p0+k]  (sampled on the fly)
//   B[k,n] = x[b, p0+k] for all n   (columns identical -> C columns = matvec)
__global__ __launch_bounds__(256) void MultilevelDense_kernel(
    const float* __restrict__ x,       // (B, D)
    const int*   __restrict__ gid,     // (B,)
    const float* __restrict__ w_mu,    // (G, D*D)
    const float* __restrict__ w_sigma, // (G, D*D)
    const float* __restrict__ b_mu,    // (G, D)
    const float* __restrict__ b_sigma, // (G, D)
    const float* __restrict__ eps_w,   // (B, D*D)
    const float* __restrict__ eps_b,   // (B, D)
    float*       __restrict__ out)     // (B, D)
{
    __shared__ float xs[DD];

    const int b    = blockIdx.x >> 3;   // sample
    const int tg   = blockIdx.x & 7;    // u-tile group (128 rows)
    const int tid  = threadIdx.x;
    const int lane = tid & 31;
    const int wave = tid >> 5;

    // Stage this sample's x row into LDS (256 threads * float4 = 1024 floats).
    {
        const v4f* src = (const v4f*)(x + (size_t)b * DD);
        ((v4f*)xs)[tid] = src[tid];
    }
    __syncthreads();

    const int g    = gid[b];
    const int u0   = tg * 128 + wave * 16;
    const int row  = u0 + (lane & 15);  // u-row this lane feeds into A
    const int koff = (lane >> 4) * 2;   // lanes 16-31 carry K=2,3

    // Uniform SGPR bases; ONE lane-varying 32-bit offset shared by all three
    // streams (w_mu/w_sigma use g*D*D, eps_w uses b*D*D — same inner layout).
    const float* muB = w_mu    + (size_t)g * (DD * DD);
    const float* sgB = w_sigma + (size_t)g * (DD * DD);
    const float* epB = eps_w   + (size_t)b * (DD * DD);
    const int off = row * DD + koff;    // element offset (lane-varying, 32-bit)

    v8f acc = {};

    for (int p0 = 0; p0 < DD; p0 += 32) {
#pragma unroll
        for (int j = 0; j < 8; ++j) {
            const int kk = p0 + 4 * j;          // uniform; 16j bytes -> IOFFSET
            v2f mu = *(const v2f*)(muB + off + kk);
            v2f sg = *(const v2f*)(sgB + off + kk);
            // eps_w is a single-use 256 MB stream: non-temporal, keep L2 for
            // the reusable group parameters.
            v2f ep = __builtin_nontemporal_load((const v2f*)(epB + off + kk));
            v2f bx = *(const v2f*)(xs + koff + kk);
            v2f a;
            a.x = fmaf(softplus_fast(sg.x), ep.x, mu.x);
            a.y = fmaf(softplus_fast(sg.y), ep.y, mu.y);
            // 8 args: (neg_a, A, neg_b, B, c_mod, C, reuse_a, reuse_b)
            acc = __builtin_amdgcn_wmma_f32_16x16x4_f32(
                false, a, false, bx, (short)0, acc, false, false);
        }
    }

    // C layout (16x16 f32): VGPR r, lanes 0-15 -> (M=r, N=lane); lanes 16-31 -> (M=8+r).
    // Column N=0 lives in lanes 0 and 16: together they hold all 16 row results.
    if ((lane & 15) == 0) {
        const int ub = u0 + (lane >> 4) * 8;
#pragma unroll
        for (int r = 0; r < 8; ++r) {
            const int u = ub + r;
            float bias = fmaf(softplus_fast(b_sigma[(size_t)g * DD + u]),
                              eps_b[(size_t)b * DD + u],
                              b_mu[(size_t)g * DD + u]);
            float v = acc[r] + bias;
            out[(size_t)b * DD + u] = fmaxf(v, 0.0f);
        }
    }
}

extern "C" void kernel_launch(void* const* d_in, const int* in_sizes, int n_in,
                              void* d_out, int out_size, void* d_ws, size_t ws_size,
                              hipStream_t stream) {
    const float* x       = (const float*)d_in[0];
    const int*   gid     = (const int*)  d_in[1];
    const float* w_mu    = (const float*)d_in[2];
    const float* w_sigma = (const float*)d_in[3];
    const float* b_mu    = (const float*)d_in[4];
    const float* b_sigma = (const float*)d_in[5];
    const float* eps_w   = (const float*)d_in[6];
    const float* eps_b   = (const float*)d_in[7];
    float* out = (float*)d_out;

    dim3 grid(BB * 8);   // 64 samples * 8 tile groups
    dim3 block(256);     // 8 waves; wave w handles u-rows [tg*128 + w*16, +16)
    hipLaunchKernelGGL(MultilevelDense_kernel, grid, block, 0, stream,
                       x, gid, w_mu, w_sigma, b_mu, b_sigma, eps_w, eps_b, out);
}